// LossFunction_kitti_29678224015626
// MI455X (gfx1250) — compile-verified
//
#include <hip/hip_runtime.h>
#include <hip/hip_bf16.h>

typedef __attribute__((ext_vector_type(2))) float v2f;
typedef __attribute__((ext_vector_type(8))) float v8f;

#define BB 2
#define KK 256
#define KNN 32
#define NN 8192
#define MM 8192
#define MARGIN 0.1f

// ---------------------------------------------------------------------------
// Kernel A: per-point squared norms + WMMA-operand packing.
// pts layout (B,3,n) coord-major.  packed layout: v2f[(b*2+half)*npts + n],
//   half 0 = (x,y), half 1 = (z, ||p||^2)
// matching the f32 16x4 operand lane layout (lanes 0-15: K=0,1; 16-31: K=2,3).
// With A's K=3 element patched to -0.5, the WMMA computes
//   c[m][n] = g_m . p_n - 0.5*||p_n||^2   =>   P = ||g||^2 - 2c.
// ---------------------------------------------------------------------------
__global__ __launch_bounds__(256)
void prep_kernel(const float* __restrict__ pts, float* __restrict__ norms,
                 v2f* __restrict__ packed, int npts)
{
    int idx = blockIdx.x * blockDim.x + threadIdx.x;
    int total = BB * npts;
    if (idx >= total) return;
    int b = idx / npts, n = idx - b * npts;
    const float* p = pts + (size_t)b * 3 * npts;
    float x = p[n], y = p[npts + n], z = p[2 * npts + n];
    float nrm = x * x + y * y + z * z;
    norms[idx] = nrm;
    v2f lo; lo.x = x; lo.y = y;
    v2f hi; hi.x = z; hi.y = nrm;
    packed[(size_t)(b * 2 + 0) * npts + n] = lo;
    packed[(size_t)(b * 2 + 1) * npts + n] = hi;
}

// ---------------------------------------------------------------------------
// Kernel B: for each "outer" point, min over all "inner" points of squared
// distance.  One wave = 16 outer rows; inner tiled by 16.
// Hot loop: ONE b64 load + ONE V_WMMA_F32_16X16X4_F32 + 8 v_max per lane.
//   min_n (on + nn - 2 g.p) = on - 2 * max_n (g.p - 0.5*nn)
// C/D layout: VGPR r, lanes 0-15 -> row M=r, lanes 16-31 -> M=r+8, col=lane%16.
// ---------------------------------------------------------------------------
__global__ __launch_bounds__(256)
void mindist_kernel(const v2f* __restrict__ packed_outer,
                    const v2f* __restrict__ packed_inner,
                    const float* __restrict__ onorm,
                    float* __restrict__ out_min, int n_outer, int n_inner)
{
    const int b      = blockIdx.y;
    const int lane   = threadIdx.x & 31;
    const int wave   = threadIdx.x >> 5;
    const int m_base = blockIdx.x * 128 + wave * 16;   // 8 waves * 16 rows / WG

    const int half = lane >> 4;        // 0: holds K=0,1   1: holds K=2,3
    const int l15  = lane & 15;

    const v2f* opack = packed_outer + (size_t)(b * 2 + half) * n_outer;
    const v2f* ipack = packed_inner + (size_t)(b * 2 + half) * n_inner;

    // A operand: outer point i = m_base + l15; patch K=3 element to -0.5
    v2f a = opack[m_base + l15];
    a.y = half ? -0.5f : a.y;          // one v_cndmask, outside the loop

    v8f maxv;
#pragma unroll
    for (int r = 0; r < 8; ++r) maxv[r] = -3.0e38f;

    for (int t = 0; t < n_inner; t += 16) {
        const v2f bv = ipack[t + l15];   // single coalesced b64 load

        v8f c = {};
        c = __builtin_amdgcn_wmma_f32_16x16x4_f32(
                false, a, false, bv, (short)0, c, false, false);

#pragma unroll
        for (int r = 0; r < 8; ++r)
            maxv[r] = fmaxf(maxv[r], c[r]);
    }

    // max-reduce the 16 columns within each 16-lane half
#pragma unroll
    for (int off = 8; off >= 1; off >>= 1) {
#pragma unroll
        for (int r = 0; r < 8; ++r) {
            float o = __shfl_xor(maxv[r], off, 16);
            maxv[r] = fmaxf(maxv[r], o);
        }
    }

    if (l15 == 0) {
        const float* onp = onorm + (size_t)b * n_outer;
#pragma unroll
        for (int r = 0; r < 8; ++r) {
            int row = m_base + half * 8 + r;
            out_min[(size_t)b * n_outer + row] = fmaf(-2.0f, maxv[r], onp[row]);
        }
    }
}

// ---------------------------------------------------------------------------
// Kernel C: single-workgroup finalize.  Huber+sum of all mins -> gal,
// keypoint transform MSE + KNN MSE -> neighborhood consensus loss.
// ---------------------------------------------------------------------------
__device__ __forceinline__ float huber_sq(float x, float c)
{
    return (x < c) ? 0.5f * x * x : fmaf(c, x, -0.5f * c * c);
}

__global__ __launch_bounds__(256)
void finalize_kernel(const float* __restrict__ minPred, const float* __restrict__ minTgt,
                     const float* __restrict__ rot, const float* __restrict__ trans,
                     const float* __restrict__ src_kp, const float* __restrict__ tgt_kp,
                     const float* __restrict__ src_knn, const float* __restrict__ tgt_knn,
                     float* __restrict__ out)
{
    const int tid = threadIdx.x;
    const float c = MARGIN;

    float gal = 0.0f;
    for (int idx = tid; idx < BB * NN; idx += 256) gal += huber_sq(minPred[idx], c);
    for (int idx = tid; idx < BB * MM; idx += 256) gal += huber_sq(minTgt[idx], c);

    // keypoints: transformed = R @ S + t  (per batch), MSE summed
    float kp = 0.0f;
    for (int idx = tid; idx < BB * 3 * KK; idx += 256) {
        int b = idx / (3 * KK);
        int rem = idx - b * 3 * KK;
        int i2 = rem / KK;
        int k  = rem - i2 * KK;
        const float* Rb = rot + b * 9;
        const float* S  = src_kp + (size_t)b * 3 * KK;
        float tr = Rb[i2 * 3 + 0] * S[k]
                 + Rb[i2 * 3 + 1] * S[KK + k]
                 + Rb[i2 * 3 + 2] * S[2 * KK + k]
                 + trans[b * 3 + i2];
        float d = tr - tgt_kp[idx];
        kp += d * d;
    }

    float knn = 0.0f;
    for (int idx = tid; idx < BB * 3 * KK * KNN; idx += 256) {
        float d = src_knn[idx] - tgt_knn[idx];
        knn += d * d;
    }

    __shared__ float sg[256], sk[256], sn[256];
    sg[tid] = gal; sk[tid] = kp; sn[tid] = knn;
    __syncthreads();
    for (int s = 128; s > 0; s >>= 1) {
        if (tid < s) {
            sg[tid] += sg[tid + s];
            sk[tid] += sk[tid + s];
            sn[tid] += sn[tid + s];
        }
        __syncthreads();
    }
    if (tid == 0) {
        float keypoints_loss = sk[0] / (float)BB;          // mean over batch of sums
        float knn_loss       = sn[0] / (float)(BB * KNN);  // mean over (B,KNN) of sums
        out[0] = knn_loss + keypoints_loss;                // neighborhood_consensus_loss
        out[1] = sg[0];                                    // gal
    }
}

// ---------------------------------------------------------------------------
extern "C" void kernel_launch(void* const* d_in, const int* in_sizes, int n_in,
                              void* d_out, int out_size, void* d_ws, size_t ws_size,
                              hipStream_t stream)
{
    (void)in_sizes; (void)n_in; (void)out_size; (void)ws_size;

    const float* src_kp   = (const float*)d_in[0];  // (B,3,K)
    const float* tgt_kp   = (const float*)d_in[1];  // (B,3,K)
    const float* rot      = (const float*)d_in[2];  // (B,3,3)
    const float* trans    = (const float*)d_in[3];  // (B,3)
    const float* src_knn  = (const float*)d_in[4];  // (B,3,K,KNN)
    const float* tgt_knn  = (const float*)d_in[5];  // (B,3,K,KNN)
    /* d_in[6] = k (scalar), compile-time constant KNN */
    const float* preds    = (const float*)d_in[7];  // src_transformed (B,3,N)
    const float* tgts     = (const float*)d_in[8];  // tgt (B,3,M)

    float* out = (float*)d_out;

    // workspace layout (floats):
    //   normP[B*N] | normT[B*M] | minPred[B*N] | minTgt[B*M]
    //   | packedP[B*2*N v2f] | packedT[B*2*M v2f]          (~768 KB total)
    float* w       = (float*)d_ws;
    float* normP   = w;
    float* normT   = w + (size_t)BB * NN;
    float* minPred = w + (size_t)2 * BB * NN;
    float* minTgt  = w + (size_t)3 * BB * NN;
    v2f*   packP   = (v2f*)(w + (size_t)4 * BB * NN);
    v2f*   packT   = packP + (size_t)BB * 2 * NN;

    // A: norms + packing
    prep_kernel<<<(BB * NN + 255) / 256, 256, 0, stream>>>(preds, normP, packP, NN);
    prep_kernel<<<(BB * MM + 255) / 256, 256, 0, stream>>>(tgts,  normT, packT, MM);

    // B: two min-distance passes (loss_1: min over tgt per pred; loss_2: swap)
    dim3 g1(NN / 128, BB);
    mindist_kernel<<<g1, 256, 0, stream>>>(packP, packT, normP, minPred, NN, MM);
    dim3 g2(MM / 128, BB);
    mindist_kernel<<<g2, 256, 0, stream>>>(packT, packP, normT, minTgt, MM, NN);

    // C: finalize both scalars
    finalize_kernel<<<1, 256, 0, stream>>>(minPred, minTgt, rot, trans,
                                           src_kp, tgt_kp, src_knn, tgt_knn, out);
}